// Prompt_43791486550104
// MI455X (gfx1250) — compile-verified
//
#include <hip/hip_runtime.h>
#include <math.h>

typedef __attribute__((ext_vector_type(2))) float v2f;
typedef __attribute__((ext_vector_type(4))) float f4;
typedef __attribute__((ext_vector_type(8))) float v8f;

#define NQ   8192
#define D    768
#define P    64
#define PD   8
#define TOPK 4
#define EPSN 1e-12f

// workspace layout (in floats)
#define WS_KN     0         // 64*768 = 49152 : normalized keys
#define WS_S      49152     // 768            : sum of 4 selected kn rows
#define WS_LOSS   49920     // 1              : loss accumulator
#define WS_COUNTS 49924     // 64 ints        : vote histogram
#define WS_TIDX   49988     // 4 ints         : selected prompt ids

// ---------------------------------------------------------------------------
// Kernel 1: L2-normalize the 64 key rows into ws; zero histogram + loss acc.
// ---------------------------------------------------------------------------
__global__ void normalize_keys(const float* __restrict__ keys,
                               float* __restrict__ ws) {
    __shared__ float red[256];
    const int row = blockIdx.x;
    const int tid = threadIdx.x;

    float ss = 0.f;
    for (int j = tid; j < D; j += 256) {
        float x = keys[row * D + j];
        ss += x * x;
    }
    red[tid] = ss;
    __syncthreads();
    for (int s = 128; s > 0; s >>= 1) {
        if (tid < s) red[tid] += red[tid + s];
        __syncthreads();
    }
    const float inv = 1.f / fmaxf(sqrtf(red[0]), EPSN);
    for (int j = tid; j < D; j += 256)
        ws[WS_KN + row * D + j] = keys[row * D + j] * inv;

    if (row == 0) {
        int* counts = (int*)(ws + WS_COUNTS);
        if (tid < P) counts[tid] = 0;
        if (tid == P) ws[WS_LOSS] = 0.f;
    }
}

// ---------------------------------------------------------------------------
// Kernel 2: sim tile via V_WMMA_F32_16X16X4_F32, per-query top-4, vote.
// Block = 256 threads (8 waves). Wave w handles 16 queries vs all 64 keys
// using 4 WMMA accumulators (N-tiles). Tile -> LDS -> top-4 scan -> atomics.
// ---------------------------------------------------------------------------
__global__ void sim_topk_vote(const float* __restrict__ queries,
                              float* __restrict__ ws) {
    __shared__ float simlds[128 * 64];   // 32 KB
    const float* kn = ws + WS_KN;

    const int tid  = threadIdx.x;
    const int wave = tid >> 5;
    const int lane = tid & 31;
    const int half = lane >> 4;          // K-half selector (A/B f32 layout)
    const int idx  = lane & 15;          // M (for A) / N (for B) within tile
    const int qbase = blockIdx.x * 128 + wave * 16;

    v8f acc0 = {}, acc1 = {}, acc2 = {}, acc3 = {};

    const float* qrow = queries + (size_t)(qbase + idx) * D + 2 * half;
    const float* kr0  = kn + (size_t)( 0 + idx) * D + 2 * half;
    const float* kr1  = kn + (size_t)(16 + idx) * D + 2 * half;
    const float* kr2  = kn + (size_t)(32 + idx) * D + 2 * half;
    const float* kr3  = kn + (size_t)(48 + idx) * D + 2 * half;

    for (int k = 0; k < D; k += 4) {
        v2f a  = *(const v2f*)(qrow + k);
        v2f b0 = *(const v2f*)(kr0 + k);
        v2f b1 = *(const v2f*)(kr1 + k);
        v2f b2 = *(const v2f*)(kr2 + k);
        v2f b3 = *(const v2f*)(kr3 + k);
        acc0 = __builtin_amdgcn_wmma_f32_16x16x4_f32(false, a, false, b0,
                                                     (short)0, acc0, false, false);
        acc1 = __builtin_amdgcn_wmma_f32_16x16x4_f32(false, a, false, b1,
                                                     (short)0, acc1, false, false);
        acc2 = __builtin_amdgcn_wmma_f32_16x16x4_f32(false, a, false, b2,
                                                     (short)0, acc2, false, false);
        acc3 = __builtin_amdgcn_wmma_f32_16x16x4_f32(false, a, false, b3,
                                                     (short)0, acc3, false, false);
    }

    // C/D layout: VGPR r, lanes 0-15 -> M=r, lanes 16-31 -> M=r+8; N = idx.
    #pragma unroll
    for (int r = 0; r < 8; ++r) {
        const int row = wave * 16 + r + 8 * half;
        simlds[row * 64 +  0 + idx] = acc0[r];
        simlds[row * 64 + 16 + idx] = acc1[r];
        simlds[row * 64 + 32 + idx] = acc2[r];
        simlds[row * 64 + 48 + idx] = acc3[r];
    }
    __syncthreads();

    if (tid < 128) {
        const float* sv = simlds + tid * 64;
        float v0 = -3.4e38f, v1 = -3.4e38f, v2 = -3.4e38f, v3 = -3.4e38f;
        int   i0 = 0, i1 = 0, i2 = 0, i3 = 0;
        for (int n = 0; n < P; ++n) {
            const float x = sv[n];
            // strict '>' => earlier index wins ties (matches jax.lax.top_k)
            if (x > v0)      { v3=v2;i3=i2; v2=v1;i2=i1; v1=v0;i1=i0; v0=x;i0=n; }
            else if (x > v1) { v3=v2;i3=i2; v2=v1;i2=i1; v1=x;i1=n; }
            else if (x > v2) { v3=v2;i3=i2; v2=x;i2=n; }
            else if (x > v3) { v3=x;i3=n; }
        }
        int* counts = (int*)(ws + WS_COUNTS);
        atomicAdd(&counts[i0], 1);
        atomicAdd(&counts[i1], 1);
        atomicAdd(&counts[i2], 1);
        atomicAdd(&counts[i3], 1);
    }
}

// ---------------------------------------------------------------------------
// Kernel 3: top-4 of the 64-bin histogram; build s = sum of 4 selected kn rows.
// ---------------------------------------------------------------------------
__global__ void select_top_prompts(float* __restrict__ ws) {
    __shared__ int tsh[TOPK];
    const int tid = threadIdx.x;
    if (tid == 0) {
        const int* counts = (const int*)(ws + WS_COUNTS);
        int c0 = -1, c1 = -1, c2 = -1, c3 = -1;
        int i0 = 0, i1 = 0, i2 = 0, i3 = 0;
        for (int n = 0; n < P; ++n) {
            const int c = counts[n];
            if (c > c0)      { c3=c2;i3=i2; c2=c1;i2=i1; c1=c0;i1=i0; c0=c;i0=n; }
            else if (c > c1) { c3=c2;i3=i2; c2=c1;i2=i1; c1=c;i1=n; }
            else if (c > c2) { c3=c2;i3=i2; c2=c;i2=n; }
            else if (c > c3) { c3=c;i3=n; }
        }
        int* tidx = (int*)(ws + WS_TIDX);
        tidx[0] = i0; tidx[1] = i1; tidx[2] = i2; tidx[3] = i3;
        tsh[0] = i0; tsh[1] = i1; tsh[2] = i2; tsh[3] = i3;
    }
    __syncthreads();
    const float* kn = ws + WS_KN;
    for (int j = tid; j < D; j += blockDim.x) {
        ws[WS_S + j] = kn[tsh[0] * D + j] + kn[tsh[1] * D + j] +
                       kn[tsh[2] * D + j] + kn[tsh[3] * D + j];
    }
}

// ---------------------------------------------------------------------------
// Kernel 4: loss = sum_i (s . q_i) / max(||q_i||, eps).  One wave per query.
// ---------------------------------------------------------------------------
__global__ void loss_reduce(const float* __restrict__ queries,
                            float* __restrict__ ws) {
    const int wid  = blockIdx.x * (blockDim.x >> 5) + (threadIdx.x >> 5);
    const int lane = threadIdx.x & 31;
    if (wid >= NQ) return;

    const float* q = queries + (size_t)wid * D;
    const float* s = ws + WS_S;
    float dot = 0.f, nrm = 0.f;
    for (int j = lane; j < D; j += 32) {
        const float x = q[j];
        dot += s[j] * x;
        nrm += x * x;
    }
    for (int m = 16; m > 0; m >>= 1) {
        dot += __shfl_xor(dot, m, 32);
        nrm += __shfl_xor(nrm, m, 32);
    }
    if (lane == 0)
        atomicAdd(ws + WS_LOSS, dot / fmaxf(sqrtf(nrm), EPSN));
}

// ---------------------------------------------------------------------------
// Kernel 5: broadcast prompts[top4] (96 KB, L2-resident) to all 8192 queries.
// 805 MB of float4 non-temporal streaming stores -> the entire runtime.
// Also writes the final loss scalar.
// ---------------------------------------------------------------------------
__global__ void broadcast_prompts(const float* __restrict__ prompts,
                                  const float* __restrict__ ws,
                                  float* __restrict__ out) {
    const int* tidx = (const int*)(ws + WS_TIDX);
    const int t0 = tidx[0], t1 = tidx[1], t2 = tidx[2], t3 = tidx[3];

    const f4* p4 = (const f4*)prompts;
    f4*       o4 = (f4*)out;

    const size_t TOTAL4 = (size_t)NQ * TOPK * PD * D / 4;  // 50,331,648
    const int    SEL4   = TOPK * PD * D / 4;               // 6144
    const int    ROW4   = PD * D / 4;                      // 1536

    const size_t stride = (size_t)gridDim.x * blockDim.x;
    for (size_t i = (size_t)blockIdx.x * blockDim.x + threadIdx.x;
         i < TOTAL4; i += stride) {
        const int inner = (int)(i % SEL4);
        const int k = inner / ROW4;
        const int w = inner % ROW4;
        const int sel = (k == 0) ? t0 : (k == 1) ? t1 : (k == 2) ? t2 : t3;
        const f4 v = p4[(size_t)sel * ROW4 + w];
        __builtin_nontemporal_store(v, &o4[i]);
    }

    if (blockIdx.x == 0 && threadIdx.x == 0)
        out[(size_t)NQ * TOPK * PD * D] = ws[WS_LOSS] / (float)NQ;
}

// ---------------------------------------------------------------------------
extern "C" void kernel_launch(void* const* d_in, const int* in_sizes, int n_in,
                              void* d_out, int out_size, void* d_ws,
                              size_t ws_size, hipStream_t stream) {
    const float* queries = (const float*)d_in[0];  // [8192, 768]
    const float* keys    = (const float*)d_in[1];  // [64, 768]
    const float* prompts = (const float*)d_in[2];  // [64, 8, 768]
    float* out = (float*)d_out;                    // [8192*32*768] + 1 (loss)
    float* ws  = (float*)d_ws;

    normalize_keys<<<P, 256, 0, stream>>>(keys, ws);
    sim_topk_vote<<<NQ / 128, 256, 0, stream>>>(queries, ws);
    select_top_prompts<<<1, 256, 0, stream>>>(ws);
    loss_reduce<<<NQ / 8, 256, 0, stream>>>(queries, ws);
    broadcast_prompts<<<4096, 256, 0, stream>>>(prompts, ws, out);
}